// QuantumCompensationBlock_64871186039156
// MI455X (gfx1250) — compile-verified
//
#include <hip/hip_runtime.h>
#include <math.h>

#define N_QUBITS   16
#define STATE_DIM  65536
#define N_SAMPLES  64
#define PI_F       3.14159265358979323846f
#define INV_SQRT2  0.70710678118654752440f

typedef __attribute__((ext_vector_type(2))) float v2f;
typedef __attribute__((ext_vector_type(8))) float v8f;

// ---------------------------------------------------------------------------
// Fused 8-qubit RY pass over groups (G0, G0+1):
//   T1   = U_a @ T          (4 chained V_WMMA_F32_16X16X4_F32)
//   T2^T = U_b @ T1^T       (4 more; T1 transposed through wave-private LDS)
// Template fusions:
//   ENCINIT  : synthesize input tile = product-state encoding * layer-0 CRZ
//              phase (low/high 8-bit factorization via LDS tables). G0==0 only.
//   FUSE_CRZ : apply layer-1 CRZ diagonal (shift=2) at write-back
//   FUSE_MEAS: accumulate per-wave Z-expectation partials (deterministic)
// AMODE 0: theta_q = ang_src[q] (weights row); 1: theta_q = tanh(x*scale)*pi
// ---------------------------------------------------------------------------
template<int G0, int AMODE, bool ENCINIT, bool FUSE_CRZ, bool FUSE_MEAS>
__global__ void __launch_bounds__(256)
ry_pair_kernel(const float* __restrict__ ang_src,
               const float* __restrict__ x,          // ENCINIT only
               const float* __restrict__ scale,      // AMODE==1 or ENCINIT
               const float* __restrict__ wcrz_row,   // ENCINIT: layer0; FUSE_CRZ: layer1
               float* __restrict__ zpart_g,          // FUSE_MEAS only
               float* __restrict__ stR, float* __restrict__ stI)
{
    static_assert(!ENCINIT || G0 == 0, "ENCINIT assumes low-group tiling");
    const int s   = blockIdx.y;      // sample
    const int tid = threadIdx.x;     // 0..255 (8 waves)

    __shared__ float Ua[16][17], Ub[16][17];
    __shared__ float stage[8][16][17];              // wave-private transpose tile
    __shared__ float cc[8], ss[8];
    __shared__ float ph[(ENCINIT || FUSE_CRZ) ? 16 : 1];
    __shared__ float v0_[ENCINIT ? 16 : 1], v1_[ENCINIT ? 16 : 1];
    __shared__ float AL[ENCINIT ? 256 : 1], AH[ENCINIT ? 256 : 1];
    __shared__ float PL[ENCINIT ? 256 : 1], PH[ENCINIT ? 256 : 1];

    // ---- setup 1: gate half-angles, encoding vectors, CRZ half-angles ----
    if (tid < 8) {
        const int q = G0 * 4 + tid;
        float theta;
        if constexpr (AMODE == 0) {
            theta = ang_src[q];
        } else {
            const int b = s >> 3, p = s & 7, h = q >> 2, w = q & 3;
            theta = tanhf(ang_src[b * 128 + h * 32 + p * 4 + w] * scale[0]) * PI_F;
        }
        sincosf(theta * 0.5f, &ss[tid], &cc[tid]);
    }
    if constexpr (ENCINIT) {
        if (tid < 16) {
            const int q = tid;
            const int b = s >> 3, p = s & 7, h = q >> 2, w = q & 3;
            const float enc = tanhf(x[b * 128 + h * 32 + p * 4 + w] * scale[0]) * PI_F;
            float sn, c;
            sincosf(enc * 0.5f, &sn, &c);
            v0_[q] = (c - sn) * INV_SQRT2;
            v1_[q] = (c + sn) * INV_SQRT2;
            ph[q]  = wcrz_row[q] * 0.5f;     // layer-0 CRZ
        }
    } else if constexpr (FUSE_CRZ) {
        if (tid < 16) ph[tid] = wcrz_row[tid] * 0.5f;   // layer-1 CRZ
    }
    __syncthreads();

    // ---- setup 2: fused 16x16 gate matrices (+ encoding tables) ----
    {
        const int m = tid >> 4, n = tid & 15;
        float ua = 1.0f, ub = 1.0f;
        #pragma unroll
        for (int k = 0; k < 4; ++k) {
            const int mb = (m >> k) & 1, nb = (n >> k) & 1;
            ua *= (mb == nb) ? cc[k]     : (mb ? ss[k]     : -ss[k]);
            ub *= (mb == nb) ? cc[4 + k] : (mb ? ss[4 + k] : -ss[4 + k]);
        }
        Ua[m][n] = ua; Ub[m][n] = ub;
    }
    if constexpr (ENCINIT) {
        const int t = tid;                       // 8-bit pattern
        float al = 1.0f, ah = 1.0f, pl = 0.0f, phh = 0.0f;
        #pragma unroll
        for (int q = 0; q < 8; ++q) {
            const int b = (t >> q) & 1;
            al *= b ? v1_[q]     : v0_[q];
            ah *= b ? v1_[q + 8] : v0_[q + 8];
        }
        #pragma unroll
        for (int q = 0; q < 7; ++q) {            // shift=1 pairs inside a byte
            const int b = (t >> q) & 1, bt = (t >> (q + 1)) & 1;
            pl  += b ? (bt ? ph[q]     : -ph[q])     : 0.0f;
            phh += b ? (bt ? ph[q + 8] : -ph[q + 8]) : 0.0f;
        }
        AL[t] = al; AH[t] = ah; PL[t] = pl; PH[t] = phh;
    }
    __syncthreads();

    const int lane = tid & 31;
    const int wave = tid >> 5;                        // 0..7
    const int j    = lane & 15;                       // lane column
    const int half = lane >> 4;
    const int o    = blockIdx.x * 8 + wave;           // other-bits combo 0..255
    constexpr int SA = 1 << (4 * G0);                 // group-a nibble stride
    constexpr int SB = SA << 4;                       // group-b nibble stride
    const int lbase   = G0 ? o : (o << 8);
    const size_t base = (size_t)s * STATE_DIM + lbase;

    // A fragments straight from LDS (computed ds addresses)
    v2f aA[4], aB[4];
    #pragma unroll
    for (int si = 0; si < 4; ++si) {
        const int k0 = 4 * si + 2 * half;
        aA[si].x = Ua[j][k0]; aA[si].y = Ua[j][k0 + 1];
        aB[si].x = Ub[j][k0]; aB[si].y = Ub[j][k0 + 1];
    }

    // ENCINIT: synthesize this lane's input-tile elements analytically.
    float bRe[8], bIe[8];
    if constexpr (ENCINIT) {
        const float ah  = AH[o];                 // hi byte == o (wave-uniform)
        const float phh = PH[o];
        const int   b8  = o & 1;                 // idx bit 8
        const int   b15 = (o >> 7) & 1;          // idx bit 15
        const float C7  = b8 ? ph[7] : -ph[7];   // gated by idx bit 7
        #pragma unroll
        for (int si = 0; si < 4; ++si) {
            #pragma unroll
            for (int e = 0; e < 2; ++e) {
                const int ia = 4 * si + 2 * half + e;
                const int lo = ia + 16 * j;      // idx low byte (SA=1, SB=16)
                const float amp = AL[lo] * ah;
                float phi = PL[lo] + phh;
                phi += ((lo >> 7) & 1) ? C7 : 0.0f;
                if (b15) phi += (lo & 1) ? ph[15] : -ph[15];   // pair (15 -> 0)
                float sn, c;
                sincosf(phi, &sn, &c);
                bRe[2 * si + e] = amp * c;
                bIe[2 * si + e] = amp * sn;
            }
        }
    }

    v8f d2r, d2i;
    #pragma unroll
    for (int plane = 0; plane < 2; ++plane) {
        // chain 1: B[k][n] = T[ia=k][ib=j]
        v8f acc = {};
        #pragma unroll
        for (int si = 0; si < 4; ++si) {
            v2f b;
            if constexpr (ENCINIT) {
                b.x = plane ? bIe[2 * si]     : bRe[2 * si];
                b.y = plane ? bIe[2 * si + 1] : bRe[2 * si + 1];
            } else {
                const float* st = plane ? stI : stR;
                const int n0 = 4 * si + 2 * half;
                if constexpr (G0 == 0) {         // contiguous pair -> b64 load
                    b = *(const v2f*)(st + base + n0 + 16 * j);
                } else {
                    b.x = st[base + (size_t)n0 * SA + (size_t)j * SB];
                    b.y = st[base + (size_t)(n0 + 1) * SA + (size_t)j * SB];
                }
            }
            acc = __builtin_amdgcn_wmma_f32_16x16x4_f32(
                false, aA[si], false, b, (short)0, acc, false, false);
        }

        // transpose T1 through wave-private LDS tile; same-wave DS ops are
        // in-order (ISA 7.1), only compiler code-motion must be fenced.
        #pragma unroll
        for (int v = 0; v < 8; ++v) stage[wave][v + 8 * half][j] = acc[v];
        __builtin_amdgcn_wave_barrier();

        // chain 2: B2[k][n] = T1[ia=n][ib=k] -> row j of staged tile
        v8f acc2 = {};
        #pragma unroll
        for (int si = 0; si < 4; ++si) {
            const int k0 = 4 * si + 2 * half;
            v2f b2;
            b2.x = stage[wave][j][k0];
            b2.y = stage[wave][j][k0 + 1];
            acc2 = __builtin_amdgcn_wmma_f32_16x16x4_f32(
                false, aB[si], false, b2, (short)0, acc2, false, false);
        }
        __builtin_amdgcn_wave_barrier();

        if (plane) d2i = acc2; else d2r = acc2;
    }

    // write-back: lane reg v holds T2[ia=j][ib=v+8*half]
    float zp[16];
    if constexpr (FUSE_MEAS) {
        #pragma unroll
        for (int k = 0; k < 16; ++k) zp[k] = 0.0f;
    }

    #pragma unroll
    for (int v = 0; v < 8; ++v) {
        const int ib = v + 8 * half;
        const int li = lbase + j * SA + ib * SB;     // index within sample
        const size_t idx = base + (size_t)j * SA + (size_t)ib * SB;
        float r = d2r[v], im = d2i[v];

        if constexpr (FUSE_CRZ) {                    // layer-1 CRZ, shift = 2
            float phi = 0.0f;
            #pragma unroll
            for (int q = 0; q < 16; ++q) {
                const int bq = (li >> q) & 1;
                const int bt = (li >> ((q + 2) & 15)) & 1;
                const float contrib = bt ? ph[q] : -ph[q];
                phi += bq ? contrib : 0.0f;
            }
            float sn, c;
            sincosf(phi, &sn, &c);
            const float nr = r * c - im * sn;
            const float ni = r * sn + im * c;
            r = nr; im = ni;
        }

        stR[idx] = r;
        stI[idx] = im;

        if constexpr (FUSE_MEAS) {
            const float pp = r * r + im * im;
            #pragma unroll
            for (int k = 0; k < 16; ++k)
                zp[k] += ((li >> k) & 1) ? -pp : pp;
        }
    }

    if constexpr (FUSE_MEAS) {
        // deterministic: per-wave partials to global, reduced in fixed order
        #pragma unroll
        for (int k = 0; k < 16; ++k) {
            float vsum = zp[k];
            #pragma unroll
            for (int off = 16; off > 0; off >>= 1)
                vsum += __shfl_down(vsum, off, 32);
            if (lane == 0)
                zpart_g[((size_t)s * 256 + o) * 16 + k] = vsum;
        }
    }
}

// ---------------------------------------------------------------------------
// Reduce per-wave Z partials (fixed order), clip, patch-transpose scatter.
// ---------------------------------------------------------------------------
__global__ void finalize_kernel(const float* __restrict__ zpart_g,
                                float* __restrict__ out)
{
    const int s = blockIdx.x;     // 64
    const int q = threadIdx.x;    // 16
    float zv = 0.0f;
    for (int w = 0; w < 256; ++w)
        zv += zpart_g[((size_t)s * 256 + w) * 16 + q];
    zv = fminf(1.0f, fmaxf(-1.0f, zv));
    const int b = s >> 3, p = s & 7, h = q >> 2, w = q & 3;
    out[b * 128 + h * 32 + p * 4 + w] = zv;
}

// ---------------------------------------------------------------------------
extern "C" void kernel_launch(void* const* d_in, const int* in_sizes, int n_in,
                              void* d_out, int out_size, void* d_ws, size_t ws_size,
                              hipStream_t stream) {
    (void)in_sizes; (void)n_in; (void)out_size; (void)ws_size;

    const float* x     = (const float*)d_in[0];  // (8,128)
    const float* wcrz  = (const float*)d_in[1];  // (2,16)
    const float* wry   = (const float*)d_in[2];  // (2,16)
    const float* scale = (const float*)d_in[3];  // scalar
    float* out = (float*)d_out;

    float* stR   = (float*)d_ws;                                   // 16 MB
    float* stI   = stR + (size_t)N_SAMPLES * STATE_DIM;            // 16 MB
    float* zpart = stI + (size_t)N_SAMPLES * STATE_DIM;            // 1 MB

    const dim3 gb(256), gg(32, N_SAMPLES);   // 8 waves/block * 32 = 256 tiles

    // K1: encoding + layer-0 CRZ synthesized in-kernel, then RY(w0) groups 0,1
    ry_pair_kernel<0, 0, true, false, false><<<gg, gb, 0, stream>>>(
        wry, x, scale, wcrz, nullptr, stR, stI);
    // K2: RY(w0) groups 2,3
    ry_pair_kernel<2, 0, false, false, false><<<gg, gb, 0, stream>>>(
        wry, nullptr, nullptr, nullptr, nullptr, stR, stI);
    // K3: RY(enc) groups 0,1
    ry_pair_kernel<0, 1, false, false, false><<<gg, gb, 0, stream>>>(
        x, nullptr, scale, nullptr, nullptr, stR, stI);
    // K4: RY(enc) groups 2,3 + layer-1 CRZ diagonal (shift=2)
    ry_pair_kernel<2, 1, false, true, false><<<gg, gb, 0, stream>>>(
        x, nullptr, scale, wcrz + 16, nullptr, stR, stI);
    // K5: RY(w1) groups 0,1
    ry_pair_kernel<0, 0, false, false, false><<<gg, gb, 0, stream>>>(
        wry + 16, nullptr, nullptr, nullptr, nullptr, stR, stI);
    // K6: RY(w1) groups 2,3 + fused Z-expectation measurement
    ry_pair_kernel<2, 0, false, false, true><<<gg, gb, 0, stream>>>(
        wry + 16, nullptr, nullptr, nullptr, zpart, stR, stI);

    // reduce partials, clip, scatter to (8,128) layout
    finalize_kernel<<<N_SAMPLES, 16, 0, stream>>>(zpart, out);
}